// Attention_30116310680398
// MI455X (gfx1250) — compile-verified
//
#include <hip/hip_runtime.h>
#include <hip/hip_bf16.h>
#include <math.h>

// Problem constants (from reference)
#define SLEN    1024
#define BATCH   64
#define DEC_HID 1024
#define ALIGN_D 1024
#define ENC_HID 2048

// slen split for the attend reduction (concurrency for the 512 MB stream)
#define SSPLIT  8
#define SCHUNK  (SLEN / SSPLIT)   // 128

typedef float v2f __attribute__((ext_vector_type(2)));
typedef float v8f __attribute__((ext_vector_type(8)));

// ---------------------------------------------------------------------------
// Kernel 1: dec_proj = s_tm1 @ sa_w^T + sa_b      (64 x 1024, K = 1024)
// One wave per 16x16 output tile; 256 tiles; K-loop of v_wmma_f32_16x16x4_f32.
// Fragment layouts per ISA 7.12.2 (f32):
//   A: lane(l16, half) holds {A[l16, k0], A[l16, k0+1]}, k0 = k + half*2
//   B: lane(l16, half) holds {B[k0, l16], B[k0+1, l16]} = sa_w[l16, k0..k0+1]
//   C/D: VGPR v -> row = v + half*8, col = l16
// ---------------------------------------------------------------------------
__global__ __launch_bounds__(32) void decproj_wmma(
    const float* __restrict__ s_tm1,   // (BATCH, DEC_HID)
    const float* __restrict__ sa_w,    // (ALIGN_D, DEC_HID)
    const float* __restrict__ sa_b,    // (ALIGN_D)
    float* __restrict__ dec_proj) {    // (BATCH, ALIGN_D)
  const int tile = blockIdx.x;        // 0..255
  const int tm   = tile >> 6;         // 0..3   (BATCH/16)
  const int tn   = tile & 63;         // 0..63  (ALIGN_D/16)
  const int lane = threadIdx.x;       // 0..31
  const int half = lane >> 4;         // 0 or 1
  const int l16  = lane & 15;

  const int rowA = tm * 16 + l16;     // batch row for A fragment
  const int rowB = tn * 16 + l16;     // align row of sa_w for B fragment
  const int col  = tn * 16 + l16;     // output column

  v8f c = {};
  for (int k = 0; k < DEC_HID; k += 4) {
    const int k0 = k + half * 2;
    v2f a = *(const v2f*)(s_tm1 + rowA * DEC_HID + k0);
    v2f b = *(const v2f*)(sa_w  + rowB * DEC_HID + k0);
    // 8 args: (neg_a, A, neg_b, B, c_mod, C, reuse_a, reuse_b)
    c = __builtin_amdgcn_wmma_f32_16x16x4_f32(
        false, a, false, b, (short)0, c, false, false);
  }
  const float bias = sa_b[col];
#pragma unroll
  for (int v = 0; v < 8; ++v) {
    const int row = tm * 16 + v + half * 8;
    dec_proj[row * ALIGN_D + col] = c[v] + bias;
  }
}

// ---------------------------------------------------------------------------
// Kernel 2: e_un[s,b] = exp( dot(tanh(dec_proj[b,:] + uh[s,b,:]), a1_w) + a1_b )
//                       * xs_mask[s,b]
// One wave per (s,b) pair; 8 waves per block; 65536 waves total.
// Streams uh (256 MB) exactly once with coalesced float4 loads.
// ---------------------------------------------------------------------------
__global__ __launch_bounds__(256) void score_kernel(
    const float* __restrict__ uh,        // (SLEN, BATCH, ALIGN_D)
    const float* __restrict__ dec_proj,  // (BATCH, ALIGN_D)
    const float* __restrict__ a1_w,      // (ALIGN_D)
    const float* __restrict__ a1_b,      // (1)
    const float* __restrict__ xs_mask,   // (SLEN, BATCH)
    float* __restrict__ e_un) {          // flat index s*BATCH + b
  const int wave = threadIdx.x >> 5;
  const int lane = threadIdx.x & 31;
  const int sb   = blockIdx.x * 8 + wave;   // 0 .. SLEN*BATCH-1
  const int b    = sb & (BATCH - 1);

  const float* __restrict__ uhp = uh + (size_t)sb * ALIGN_D;
  const float* __restrict__ dpp = dec_proj + (size_t)b * ALIGN_D;

  float acc = 0.f;
#pragma unroll
  for (int j = 0; j < ALIGN_D / 128; ++j) {   // 8 iters x 128 floats per wave
    const int idx = j * 128 + lane * 4;
    float4 u = *(const float4*)(uhp + idx);
    float4 d = *(const float4*)(dpp + idx);
    float4 w = *(const float4*)(a1_w + idx);
    acc = fmaf(tanhf(u.x + d.x), w.x, acc);
    acc = fmaf(tanhf(u.y + d.y), w.y, acc);
    acc = fmaf(tanhf(u.z + d.z), w.z, acc);
    acc = fmaf(tanhf(u.w + d.w), w.w, acc);
  }
  // wave32 butterfly reduction
#pragma unroll
  for (int off = 16; off > 0; off >>= 1) acc += __shfl_xor(acc, off, 32);

  if (lane == 0) {
    const float score = acc + a1_b[0];
    e_un[sb] = __expf(score) * xs_mask[sb];
  }
}

// ---------------------------------------------------------------------------
// Kernel 3: colsum[b] = sum_s e_un[s,b]
// ---------------------------------------------------------------------------
__global__ __launch_bounds__(256) void colsum_kernel(
    const float* __restrict__ e_un, float* __restrict__ colsum) {
  const int b = blockIdx.x;
  float acc = 0.f;
  for (int s = threadIdx.x; s < SLEN; s += 256)
    acc += e_un[s * BATCH + b];
  __shared__ float red[256];
  red[threadIdx.x] = acc;
  __syncthreads();
  for (int off = 128; off > 0; off >>= 1) {
    if (threadIdx.x < off) red[threadIdx.x] += red[threadIdx.x + off];
    __syncthreads();
  }
  if (threadIdx.x == 0) colsum[b] = red[0];
}

// ---------------------------------------------------------------------------
// Kernel 4: e_ij[s,b] = e_un[s,b] / colsum[b]   (first output)
// ---------------------------------------------------------------------------
__global__ __launch_bounds__(256) void eij_kernel(
    const float* __restrict__ e_un, const float* __restrict__ colsum,
    float* __restrict__ e_ij) {
  const int i = blockIdx.x * 256 + threadIdx.x;
  e_ij[i] = e_un[i] / colsum[i & (BATCH - 1)];
}

// ---------------------------------------------------------------------------
// Kernel 5a: partial[c,b,e] = sum_{s in chunk c} w[s,b] * xs_h[s,b,e]
// grid = (BATCH, SSPLIT) = 512 blocks (4096 waves) so the 512 MB xs_h stream
// has enough loads in flight to reach HBM bandwidth. Each thread keeps two
// float4 accumulators (e and e+1024) for extra memory-level parallelism.
// ---------------------------------------------------------------------------
__global__ __launch_bounds__(256) void attend_partial_kernel(
    const float* __restrict__ xs_h,    // (SLEN, BATCH, ENC_HID)
    const float* __restrict__ e_un,    // (SLEN, BATCH)
    const float* __restrict__ colsum,  // (BATCH)
    float* __restrict__ partial) {     // (SSPLIT, BATCH, ENC_HID)
  const int b     = blockIdx.x;
  const int chunk = blockIdx.y;
  const int s0    = chunk * SCHUNK;

  __shared__ float w[SCHUNK];
  const float inv = 1.f / colsum[b];
  for (int s = threadIdx.x; s < SCHUNK; s += 256)
    w[s] = e_un[(s0 + s) * BATCH + b] * inv;
  __syncthreads();

  const int e0 = threadIdx.x * 4;        // 0..1020
  const int e1 = e0 + 1024;              // 1024..2044
  float4 acc0 = make_float4(0.f, 0.f, 0.f, 0.f);
  float4 acc1 = make_float4(0.f, 0.f, 0.f, 0.f);

  const float*  base   = xs_h + ((size_t)s0 * BATCH + b) * ENC_HID;
  const size_t  stride = (size_t)BATCH * ENC_HID;
#pragma unroll 4
  for (int s = 0; s < SCHUNK; ++s) {
    const float* row = base + (size_t)s * stride;
    float4 x0 = *(const float4*)(row + e0);
    float4 x1 = *(const float4*)(row + e1);
    const float ws = w[s];
    acc0.x = fmaf(ws, x0.x, acc0.x);
    acc0.y = fmaf(ws, x0.y, acc0.y);
    acc0.z = fmaf(ws, x0.z, acc0.z);
    acc0.w = fmaf(ws, x0.w, acc0.w);
    acc1.x = fmaf(ws, x1.x, acc1.x);
    acc1.y = fmaf(ws, x1.y, acc1.y);
    acc1.z = fmaf(ws, x1.z, acc1.z);
    acc1.w = fmaf(ws, x1.w, acc1.w);
  }
  float* p = partial + ((size_t)chunk * BATCH + b) * ENC_HID;
  *(float4*)(p + e0) = acc0;
  *(float4*)(p + e1) = acc1;
}

// ---------------------------------------------------------------------------
// Kernel 5b: attend[b,e] = sum_c partial[c,b,e]   (fixed order, deterministic)
// ---------------------------------------------------------------------------
__global__ __launch_bounds__(256) void attend_reduce_kernel(
    const float* __restrict__ partial, float* __restrict__ attend) {
  const size_t i = ((size_t)blockIdx.x * 256 + threadIdx.x) * 4;
  float4 acc = make_float4(0.f, 0.f, 0.f, 0.f);
#pragma unroll
  for (int c = 0; c < SSPLIT; ++c) {
    float4 x = *(const float4*)(partial + (size_t)c * BATCH * ENC_HID + i);
    acc.x += x.x; acc.y += x.y; acc.z += x.z; acc.w += x.w;
  }
  *(float4*)(attend + i) = acc;
}

// ---------------------------------------------------------------------------
// Fallback (only if workspace is too small for partials): unsplit attend.
// ---------------------------------------------------------------------------
__global__ __launch_bounds__(256) void attend_kernel(
    const float* __restrict__ xs_h, const float* __restrict__ e_un,
    const float* __restrict__ colsum, float* __restrict__ attend) {
  const int b  = blockIdx.x;
  const int e0 = blockIdx.y * 1024 + threadIdx.x * 4;

  __shared__ float w[SLEN];
  const float inv = 1.f / colsum[b];
  for (int s = threadIdx.x; s < SLEN; s += 256)
    w[s] = e_un[s * BATCH + b] * inv;
  __syncthreads();

  float4 acc = make_float4(0.f, 0.f, 0.f, 0.f);
  const float* base = xs_h + (size_t)b * ENC_HID + e0;
  const size_t stride = (size_t)BATCH * ENC_HID;
#pragma unroll 8
  for (int s = 0; s < SLEN; ++s) {
    float4 x = *(const float4*)(base + (size_t)s * stride);
    const float ws = w[s];
    acc.x = fmaf(ws, x.x, acc.x);
    acc.y = fmaf(ws, x.y, acc.y);
    acc.z = fmaf(ws, x.z, acc.z);
    acc.w = fmaf(ws, x.w, acc.w);
  }
  *(float4*)(attend + (size_t)b * ENC_HID + e0) = acc;
}

// ---------------------------------------------------------------------------
// Launch
// ---------------------------------------------------------------------------
extern "C" void kernel_launch(void* const* d_in, const int* in_sizes, int n_in,
                              void* d_out, int out_size, void* d_ws, size_t ws_size,
                              hipStream_t stream) {
  const float* s_tm1   = (const float*)d_in[0];  // (B, DEC_HID)
  const float* xs_h    = (const float*)d_in[1];  // (SLEN, B, ENC_HID)
  const float* uh      = (const float*)d_in[2];  // (SLEN, B, ALIGN)
  const float* xs_mask = (const float*)d_in[3];  // (SLEN, B)
  const float* sa_w    = (const float*)d_in[4];  // (ALIGN, DEC_HID)
  const float* sa_b    = (const float*)d_in[5];  // (ALIGN)
  const float* a1_w    = (const float*)d_in[6];  // (1, ALIGN)
  const float* a1_b    = (const float*)d_in[7];  // (1)

  float* out    = (float*)d_out;
  float* e_ij   = out;                        // (SLEN, B)   = 65536 floats
  float* attend = out + SLEN * BATCH;         // (B, ENC_HID) = 131072 floats

  float* ws       = (float*)d_ws;
  float* dec_proj = ws;                       // 65536 floats
  float* e_un     = ws + BATCH * ALIGN_D;     // 65536 floats
  float* colsum   = e_un + SLEN * BATCH;      // 64 floats
  float* partial  = colsum + 64;              // SSPLIT*BATCH*ENC_HID = 1M floats

  const size_t need_bytes =
      (size_t)(BATCH * ALIGN_D + SLEN * BATCH + 64 +
               SSPLIT * BATCH * ENC_HID) * sizeof(float);

  // 1) dec_proj GEMM via WMMA f32 16x16x4
  decproj_wmma<<<dim3(256), dim3(32), 0, stream>>>(s_tm1, sa_w, sa_b, dec_proj);

  // 2) fused add+tanh+dot+exp+mask over uh (256 MB stream)
  score_kernel<<<dim3((SLEN * BATCH) / 8), dim3(256), 0, stream>>>(
      uh, dec_proj, a1_w, a1_b, xs_mask, e_un);

  // 3) per-column sums
  colsum_kernel<<<dim3(BATCH), dim3(256), 0, stream>>>(e_un, colsum);

  // 4) normalized attention weights (output 0)
  eij_kernel<<<dim3((SLEN * BATCH) / 256), dim3(256), 0, stream>>>(e_un, colsum, e_ij);

  // 5) attended context (output 1): split-slen for bandwidth, then reduce
  if (ws_size >= need_bytes) {
    attend_partial_kernel<<<dim3(BATCH, SSPLIT), dim3(256), 0, stream>>>(
        xs_h, e_un, colsum, partial);
    attend_reduce_kernel<<<dim3(BATCH * ENC_HID / 1024), dim3(256), 0, stream>>>(
        partial, attend);
  } else {
    attend_kernel<<<dim3(BATCH, ENC_HID / 1024), dim3(256), 0, stream>>>(
        xs_h, e_un, colsum, attend);
  }
}